// TBSyntaxParser_24275155157382
// MI455X (gfx1250) — compile-verified
//
#include <hip/hip_runtime.h>
#include <hip/hip_bf16.h>

typedef float v2f __attribute__((ext_vector_type(2)));
typedef float v8f __attribute__((ext_vector_type(8)));

#define BSTATES 4096
#define LBUF    128
#define DW      60
#define KX      360     // 6*D
#define XSTR    364     // LDS row stride in floats (1456 B, multiple of 8)
#define HID     200
#define HIDP    208     // 13*16 padded hidden
#define OUTN    3
#define NT1     13      // n-tiles in layer 1
#define KB1     90      // K blocks of 4 in layer 1 (360/4)
#define KB2     52      // K blocks of 4 in layer 2 (208/4)

// ---- Kernel 0: pre-swizzle W1/W2 into exact WMMA B-fragment order ----------
// w1f: [NT1][KB1][32 lanes] x v2f   (cols >= 200 zeroed)
// w2f: [KB2][32 lanes] x v2f       (cols >= 3 zeroed, K >= 200 zeroed)
__global__ __launch_bounds__(256) void pack_w_frags(
    const float* __restrict__ W1, const float* __restrict__ W2,
    float* __restrict__ w1f, float* __restrict__ w2f)
{
  const int NW1 = NT1 * KB1 * 32;
  const int NW2 = KB2 * 32;
  int i = blockIdx.x * blockDim.x + threadIdx.x;
  if (i < NW1) {
    int lane = i & 31;
    int kb   = (i >> 5) % KB1;
    int nt   = (i >> 5) / KB1;
    int n    = nt * 16 + (lane & 15);
    int k    = kb * 4 + ((lane >> 4) << 1);
    w1f[2 * i + 0] = (n < HID) ? W1[k * HID + n]       : 0.0f;
    w1f[2 * i + 1] = (n < HID) ? W1[(k + 1) * HID + n] : 0.0f;
  } else if (i < NW1 + NW2) {
    int j    = i - NW1;
    int lane = j & 31;
    int kb   = j >> 5;
    int n    = lane & 15;
    int k    = kb * 4 + ((lane >> 4) << 1);
    w2f[2 * j + 0] = (n < OUTN && k < HID)     ? W2[k * OUTN + n]       : 0.0f;
    w2f[2 * j + 1] = (n < OUTN && k + 1 < HID) ? W2[(k + 1) * OUTN + n] : 0.0f;
  }
}

// ---------------- Kernel 1: hid = relu(gather(X) @ W1 + b1) ----------------
// block = 416 threads = 13 waves; each wave owns one 16-col slice of HIDP.
__global__ __launch_bounds__(416) void parser_layer1(
    const float* __restrict__ buffer, const float* __restrict__ w1f,
    const float* __restrict__ b1, const int* __restrict__ bidx,
    const int* __restrict__ sidx, float* __restrict__ hid)
{
  __shared__ float xs[16 * XSTR];
  __shared__ int   srow[16 * 6];

  const int t  = threadIdx.x;
  const int m0 = blockIdx.x * 16;

  // resolve the 6 source rows per state once
  if (t < 96) {
    int r = t / 6, s = t - r * 6;
    int m = m0 + r;
    srow[t] = (s < 3) ? (bidx[m] + s) : sidx[m * 3 + (s - 3)];
  }
  __syncthreads();

  // cooperative gather of the 16x360 X tile into LDS
  for (int e = t; e < 16 * KX; e += 416) {
    int r = e / KX, k = e - r * KX;
    int s = k / DW, off = k - s * DW;
    int m = m0 + r;
    xs[r * XSTR + k] =
        buffer[((size_t)m * LBUF + srow[r * 6 + s]) * DW + off];
  }
  __syncthreads();

  const int lane = t & 31;
  const int wave = t >> 5;                   // 0..12 -> n-tile
  const int col  = wave * 16 + (lane & 15);
  const int koff = ((lane >> 4) & 1) * 2;    // upper half-wave holds K+2,K+3
  const int arow = lane & 15;

  const v2f* __restrict__ bfrag =
      (const v2f*)w1f + (size_t)wave * KB1 * 32 + lane;

  v8f c = {};
  #pragma unroll 5
  for (int kb = 0; kb < KB1; ++kb) {
    v2f a = *(const v2f*)&xs[arow * XSTR + kb * 4 + koff];  // ds_load_b64
    v2f b = bfrag[kb * 32];                                 // coalesced b64
    c = __builtin_amdgcn_wmma_f32_16x16x4_f32(false, a, false, b,
                                              (short)0, c, false, false);
  }

  const float bias = (col < HID) ? b1[col] : 0.0f;
  #pragma unroll
  for (int r = 0; r < 8; ++r) {
    int row = m0 + r + ((lane >> 4) << 3);
    float v = (col < HID) ? fmaxf(c[r] + bias, 0.0f) : 0.0f;  // pad cols -> 0
    hid[(size_t)row * HIDP + col] = v;
  }
}

// ---------------- Kernel 2: out = hid @ W2 + b2 ; legal mask ----------------
// block = 128 = 4 waves; each wave one 16-row tile.
__global__ __launch_bounds__(128) void parser_layer2(
    const float* __restrict__ hid, const float* __restrict__ w2f,
    const float* __restrict__ b2, const int* __restrict__ bidx,
    const int* __restrict__ slen, float* __restrict__ out,
    float* __restrict__ legal)
{
  const int t    = threadIdx.x;
  const int lane = t & 31;
  const int wave = t >> 5;
  const int m0   = (blockIdx.x * 4 + wave) * 16;
  const int col  = lane & 15;
  const int koff = ((lane >> 4) & 1) * 2;
  const int arow = m0 + (lane & 15);

  const v2f* __restrict__ bfrag = (const v2f*)w2f + lane;

  v8f c = {};
  #pragma unroll 4
  for (int kb = 0; kb < KB2; ++kb) {      // hid pad region is 0, w2f padded
    v2f a = *(const v2f*)&hid[(size_t)arow * HIDP + kb * 4 + koff];
    v2f b = bfrag[kb * 32];
    c = __builtin_amdgcn_wmma_f32_16x16x4_f32(false, a, false, b,
                                              (short)0, c, false, false);
  }

  if (col < OUTN) {
    const float bias = b2[col];
    #pragma unroll
    for (int r = 0; r < 8; ++r) {
      int row = m0 + r + ((lane >> 4) << 3);
      out[row * OUTN + col] = c[r] + bias;
    }
  }

  // legality mask: lanes 0..15 each handle one state
  if (lane < 16) {
    int m  = m0 + lane;
    int bi = bidx[m];
    int sl = slen[m];
    legal[m * 3 + 0] = (bi + 3 >= LBUF) ? 0.0f : 1.0f;  // shift
    legal[m * 3 + 1] = (sl <= 3)        ? 0.0f : 1.0f;  // left-arc
    legal[m * 3 + 2] = (sl <= 4)        ? 0.0f : 1.0f;  // right-arc
  }
}

extern "C" void kernel_launch(void* const* d_in, const int* in_sizes, int n_in,
                              void* d_out, int out_size, void* d_ws, size_t ws_size,
                              hipStream_t stream) {
  const float* buffer = (const float*)d_in[0];
  const float* W1     = (const float*)d_in[1];
  const float* b1     = (const float*)d_in[2];
  const float* W2     = (const float*)d_in[3];
  const float* b2     = (const float*)d_in[4];
  const int*   bidx   = (const int*)d_in[5];
  const int*   sidx   = (const int*)d_in[6];
  const int*   slen   = (const int*)d_in[7];

  float* hid = (float*)d_ws;                          // [4096,208] ~3.33 MB
  float* w1f = hid + (size_t)BSTATES * HIDP;          // 13*90*32*2 floats
  float* w2f = w1f + (size_t)NT1 * KB1 * 32 * 2;      // 52*32*2 floats

  float* out   = (float*)d_out;                       // [4096, 3]
  float* legal = out + (size_t)BSTATES * OUTN;

  const int packN = NT1 * KB1 * 32 + KB2 * 32;
  pack_w_frags<<<(packN + 255) / 256, 256, 0, stream>>>(W1, W2, w1f, w2f);
  parser_layer1<<<BSTATES / 16, 416, 0, stream>>>(buffer, w1f, b1, bidx, sidx, hid);
  parser_layer2<<<BSTATES / (16 * 4), 128, 0, stream>>>(hid, w2f, b2, bidx, slen,
                                                        out, legal);
}